// DiffusionSdeOptimal_37769942401171
// MI455X (gfx1250) — compile-verified
//
#include <hip/hip_runtime.h>
#include <hip/hip_bf16.h>
#include <float.h>

typedef float v2f __attribute__((ext_vector_type(2)));
typedef float v8f __attribute__((ext_vector_type(8)));

#define NCLS 10
#define NCLS_PAD 16
#define KCHUNK 1024   // samples per block in the WMMA pass (4 waves x 256)
#define MAXD4 768     // D/4 for D=3072

// Pass 1: inner[n] = -(0.5/sigma2) * || u - sqrt_a * x_n ||^2
// One wave32 per row; u cached in LDS; float4 streaming loads of x.
__global__ void k_inner(const float* __restrict__ u, const float* __restrict__ x,
                        const float* __restrict__ alpha_bar, const int* __restrict__ t,
                        float* __restrict__ inner, int N, int D) {
  __shared__ float4 su[MAXD4];
  const int D4 = D >> 2;
  const float a_bar = alpha_bar[*t];
  const float s = sqrtf(a_bar);
  const float scale = -0.5f / (1.0f - a_bar);
  const float4* u4 = (const float4*)u;
  for (int i = threadIdx.x; i < D4; i += blockDim.x) su[i] = u4[i];
  __syncthreads();
  const int wave = threadIdx.x >> 5;
  const int lane = threadIdx.x & 31;
  const int row  = blockIdx.x * (blockDim.x >> 5) + wave;
  if (row >= N) return;
  const float4* x4 = (const float4*)(x + (size_t)row * D);
  float acc = 0.0f;
  for (int i = lane; i < D4; i += 32) {
    const float4 xv = x4[i];
    const float4 uv = su[i];
    const float d0 = uv.x - s * xv.x;
    const float d1 = uv.y - s * xv.y;
    const float d2 = uv.z - s * xv.z;
    const float d3 = uv.w - s * xv.w;
    acc += d0 * d0 + d1 * d1 + d2 * d2 + d3 * d3;
  }
  #pragma unroll
  for (int off = 16; off > 0; off >>= 1) acc += __shfl_xor(acc, off, 32);
  if (lane == 0) inner[row] = scale * acc;
}

// Pass 2: per-class max + sum-exp over inner[]; also zero the combo accumulator.
__global__ void k_stats(const float* __restrict__ inner, const int* __restrict__ y,
                        float* __restrict__ stats /* m[16], Z[16] */,
                        float* __restrict__ combo, int N, int D) {
  for (int i = threadIdx.x; i < NCLS_PAD * D; i += blockDim.x) combo[i] = 0.0f;

  __shared__ float wred[32][NCLS];
  __shared__ float sm[NCLS];
  const int wave = threadIdx.x >> 5;
  const int lane = threadIdx.x & 31;
  const int nwaves = blockDim.x >> 5;

  float lm[NCLS];
  #pragma unroll
  for (int c = 0; c < NCLS; ++c) lm[c] = -FLT_MAX;
  for (int n = threadIdx.x; n < N; n += blockDim.x) {
    const int cls = y[n];
    const float v = inner[n];
    #pragma unroll
    for (int c = 0; c < NCLS; ++c) lm[c] = fmaxf(lm[c], (cls == c) ? v : -FLT_MAX);
  }
  #pragma unroll
  for (int c = 0; c < NCLS; ++c) {
    float v = lm[c];
    #pragma unroll
    for (int off = 16; off > 0; off >>= 1) v = fmaxf(v, __shfl_xor(v, off, 32));
    if (lane == 0) wred[wave][c] = v;
  }
  __syncthreads();
  if (threadIdx.x < NCLS) {
    float v = -FLT_MAX;
    for (int w = 0; w < nwaves; ++w) v = fmaxf(v, wred[w][threadIdx.x]);
    sm[threadIdx.x] = v;
    stats[threadIdx.x] = v;
  }
  __syncthreads();

  float ls[NCLS];
  #pragma unroll
  for (int c = 0; c < NCLS; ++c) ls[c] = 0.0f;
  for (int n = threadIdx.x; n < N; n += blockDim.x) {
    const int cls = y[n];
    const float e = __expf(inner[n] - sm[cls]);
    #pragma unroll
    for (int c = 0; c < NCLS; ++c) ls[c] += (cls == c) ? e : 0.0f;
  }
  #pragma unroll
  for (int c = 0; c < NCLS; ++c) {
    float v = ls[c];
    #pragma unroll
    for (int off = 16; off > 0; off >>= 1) v += __shfl_xor(v, off, 32);
    if (lane == 0) wred[wave][c] = v;
  }
  __syncthreads();
  if (threadIdx.x < NCLS) {
    float v = 0.0f;
    for (int w = 0; w < nwaves; ++w) v += wred[w][threadIdx.x];
    stats[NCLS_PAD + threadIdx.x] = v;
  }
}

// Pass 3a: normalized softmax weight per sample.
__global__ void k_weights(const float* __restrict__ inner, const int* __restrict__ y,
                          const float* __restrict__ stats, float* __restrict__ p, int N) {
  const int n = blockIdx.x * blockDim.x + threadIdx.x;
  if (n < N) {
    const int c = y[n];
    p[n] = __expf(inner[n] - stats[c]) / stats[NCLS_PAD + c];
  }
}

// Pass 3b: combo(16 x D) += W(16 x N) @ X(N x D) via V_WMMA_F32_16X16X4_F32.
// grid.x = D/16 column tiles, grid.y = K-chunks; 4 waves per block.
// A tile 16x4 f32: lanes 0-15 hold K=0,1 (regs .x,.y); lanes 16-31 hold K=2,3.
// B tile 4x16 f32: mirrored (rows striped across lanes within a VGPR).
__global__ void __launch_bounds__(128)
k_combo(const float* __restrict__ x, const int* __restrict__ y,
        const float* __restrict__ p, float* __restrict__ combo,
        int N, int D) {
  const int colBase = blockIdx.x * 16;
  const int wave = threadIdx.x >> 5;
  const int lane = threadIdx.x & 31;
  const int g   = lane >> 4;   // half-wave: selects K pair (0,1) vs (2,3)
  const int idx = lane & 15;   // class row (A) / column within tile (B)
  const int samplesPerWave = KCHUNK / 4;               // 256
  const int n0 = blockIdx.y * KCHUNK + wave * samplesPerWave;

  v8f acc0 = {0.f, 0.f, 0.f, 0.f, 0.f, 0.f, 0.f, 0.f};
  v8f acc1 = {0.f, 0.f, 0.f, 0.f, 0.f, 0.f, 0.f, 0.f};

  if (n0 + samplesPerWave <= N) {
    // ---- Fast path: no bounds checks, strength-reduced addressing ----
    const float*  xp = x + (size_t)(n0 + 2 * g) * D + colBase + idx;
    const int2*   y2 = (const int2*)(y + n0);    // pairs of labels
    const float2* p2 = (const float2*)(p + n0);  // pairs of weights
    const size_t strideX = (size_t)4 * D;        // advance 4 samples per step
    #pragma unroll 4
    for (int s = 0; s < samplesPerWave / 4; ++s) {
      const int2   yv = y2[2 * s + g];   // labels  (n0+4s+2g, n0+4s+2g+1)
      const float2 pv = p2[2 * s + g];   // weights (same samples)
      v2f a, b;
      a.x = (yv.x == idx) ? pv.x : 0.f;  // v_cndmask, no branch
      a.y = (yv.y == idx) ? pv.y : 0.f;
      b.x = xp[0];
      b.y = xp[D];
      xp += strideX;
      if (s & 1)
        acc1 = __builtin_amdgcn_wmma_f32_16x16x4_f32(
            false, a, false, b, (short)0, acc1, false, false);
      else
        acc0 = __builtin_amdgcn_wmma_f32_16x16x4_f32(
            false, a, false, b, (short)0, acc0, false, false);
    }
  } else {
    // ---- Tail path (last chunk only): guarded, branchless selects ----
    for (int s = 0; s < samplesPerWave / 4; ++s) {
      const int nb  = n0 + s * 4;
      const int na0 = nb + 2 * g;
      const int na1 = na0 + 1;
      v2f a, b;
      a.x = 0.f; a.y = 0.f; b.x = 0.f; b.y = 0.f;
      if (na0 < N) {
        const int   y0 = y[na0];
        const float w0 = p[na0];
        a.x = (y0 == idx) ? w0 : 0.f;
        b.x = x[(size_t)na0 * D + colBase + idx];
      }
      if (na1 < N) {
        const int   y1 = y[na1];
        const float w1 = p[na1];
        a.y = (y1 == idx) ? w1 : 0.f;
        b.y = x[(size_t)na1 * D + colBase + idx];
      }
      acc0 = __builtin_amdgcn_wmma_f32_16x16x4_f32(
          false, a, false, b, (short)0, acc0, false, false);
    }
  }

  const v8f acc = acc0 + acc1;

  // Cross-wave reduction of the four 16x16 partial tiles, then one atomicAdd each.
  __shared__ float red[4][16][16];
  #pragma unroll
  for (int r = 0; r < 8; ++r) red[wave][r + 8 * g][idx] = acc[r];
  __syncthreads();
  for (int e = threadIdx.x; e < 256; e += blockDim.x) {
    const int m = e >> 4;
    const int n = e & 15;
    const float v = red[0][m][n] + red[1][m][n] + red[2][m][n] + red[3][m][n];
    atomicAdd(&combo[(size_t)m * D + colBase + n], v);
  }
}

// Pass 4: out[c,d] = -(1/sigma2) * (u[d] - sqrt_a * combo[c,d]) for c < 10.
__global__ void k_final(const float* __restrict__ u, const float* __restrict__ combo,
                        const float* __restrict__ alpha_bar, const int* __restrict__ t,
                        float* __restrict__ out, int D) {
  const int i = blockIdx.x * blockDim.x + threadIdx.x;
  if (i >= NCLS * D) return;
  const int c = i / D;
  const int d = i - c * D;
  const float a_bar = alpha_bar[*t];
  const float s = sqrtf(a_bar);
  const float inv = -1.0f / (1.0f - a_bar);
  out[i] = inv * (u[d] - s * combo[(size_t)c * D + d]);
}

extern "C" void kernel_launch(void* const* d_in, const int* in_sizes, int n_in,
                              void* d_out, int out_size, void* d_ws, size_t ws_size,
                              hipStream_t stream) {
  (void)n_in; (void)out_size; (void)ws_size;
  const float* u  = (const float*)d_in[0];   // (1,3,32,32) fp32 -> D
  const float* x  = (const float*)d_in[1];   // (N, D) fp32
  const int*   y  = (const int*)d_in[2];     // (N,) int32
  const float* ab = (const float*)d_in[3];   // (1000,) fp32
  const int*   t  = (const int*)d_in[4];     // scalar int
  float* out = (float*)d_out;

  const int D = in_sizes[0];
  const int N = in_sizes[2];

  // Workspace partition (256B-aligned): inner[N], p[N], stats[32], combo[16*D]
  char* ws = (char*)d_ws;
  float* inner = (float*)ws;
  size_t off = ((size_t)N * 4 + 255) & ~(size_t)255;
  float* p = (float*)(ws + off);
  off += ((size_t)N * 4 + 255) & ~(size_t)255;
  float* stats = (float*)(ws + off);
  off += 256;
  float* combo = (float*)(ws + off);

  const int rowsPerBlock = 256 / 32;  // 8 waves -> 8 rows
  k_inner<<<(N + rowsPerBlock - 1) / rowsPerBlock, 256, 0, stream>>>(u, x, ab, t, inner, N, D);
  k_stats<<<1, 1024, 0, stream>>>(inner, y, stats, combo, N, D);
  k_weights<<<(N + 255) / 256, 256, 0, stream>>>(inner, y, stats, p, N);
  dim3 gridCombo(D / 16, (N + KCHUNK - 1) / KCHUNK);
  k_combo<<<gridCombo, 128, 0, stream>>>(x, y, p, combo, N, D);
  k_final<<<(NCLS * D + 255) / 256, 256, 0, stream>>>(u, combo, ab, t, out, D);
}